// FairGCN_38113539785176
// MI455X (gfx1250) — compile-verified
//
#include <hip/hip_runtime.h>

// ---------------------------------------------------------------------------
// FairGCN forward for gfx1250 (MI455X).
//  - GEMMs via V_WMMA_F32_16X16X4_F32 (full f32, matches reference numerics)
//  - Row-guard hoisted to a kernel template param + separate tail launch:
//    no EXEC churn and no accumulator phi-copies in the K-loop
//  - Vectorized float4 LDS staging of weights when NCOLS == 128
//  - Edge aggregation: one wave32 per edge, global_atomic_add_f32 (L2-resident)
//  - BatchNorm training stats via LDS block reduction + per-column atomics
// ---------------------------------------------------------------------------

typedef __attribute__((ext_vector_type(2))) float v2f;
typedef __attribute__((ext_vector_type(8))) float v8f;

#define DCH 128     // feature channels
#define KC  64      // K-chunk staged in LDS

// ----------------------------- utility kernels -----------------------------

__global__ __launch_bounds__(256) void zero_kernel(float* __restrict__ p, long long n) {
    long long i = (long long)blockIdx.x * 256 + threadIdx.x;
    long long stride = (long long)gridDim.x * 256;
    for (; i < n; i += stride) p[i] = 0.0f;
}

__global__ __launch_bounds__(256) void deg_kernel(const int* __restrict__ dst,
                                                  float* __restrict__ deg, int E) {
    int e = blockIdx.x * 256 + threadIdx.x;
    if (e < E) atomicAdd(&deg[dst[e]], 1.0f);
}

__global__ __launch_bounds__(256) void dinv_kernel(float* __restrict__ deg, int n) {
    int i = blockIdx.x * 256 + threadIdx.x;
    if (i < n) deg[i] = rsqrtf(deg[i] + 1.0f);   // deg includes self-loop
}

// ----------------------------- WMMA f32 GEMM --------------------------------
// Out[M x NCOLS] = A[M x K] * W[K x NCOLS] (+bias)(+relu)
// A supplied as (A1 | A2): A1 rows give K=0..127, A2 rows give K=128..255
// (concat case); lda is always 128.
// mode: 0 = raw store, 1 = +bias then relu, 2 = +bias
// Block: 256 threads (8 waves). Wave w -> output cols [16w,16w+16).
// Block -> 64 rows (4 strips of 16). Weights staged in 32KB LDS per K-chunk.
// GUARD=false kernel covers only full 64-row blocks; GUARD=true handles the
// ragged tail (launched with blockBase = #full blocks).

template<bool GUARD>
__global__ __launch_bounds__(256) void gemm_wmma_kernel(
    const float* __restrict__ A1, const float* __restrict__ A2,
    const float* __restrict__ W,  const float* __restrict__ bias,
    float* __restrict__ Out, int M, int K, int NCOLS, int ldout, int mode,
    int blockBase)
{
    __shared__ float bs[KC * DCH];   // 32 KB

    const int lane = threadIdx.x & 31;
    const int wave = threadIdx.x >> 5;
    const int m_l  = lane & 15;                 // M index within A tile
    const int vk   = (lane >> 4) << 1;          // K sub-offset: 0 or 2
    const int ncol = (wave << 4) + (lane & 15); // 0..127 output column
    const int r0   = (blockBase + blockIdx.x) * 64;

    v8f zero8 = {0.f,0.f,0.f,0.f,0.f,0.f,0.f,0.f};
    v8f acc[4] = {zero8, zero8, zero8, zero8};

    for (int kc = 0; kc < K; kc += KC) {
        // ---- stage W[kc:kc+KC, 0:128) into LDS ----
        if (NCOLS == DCH) {
            // dense case: straight float4 copy (global_load_b128 / ds_store_b128)
            float4* dstv = (float4*)bs;
            const float4* srcv = (const float4*)(W + kc * DCH);
            #pragma unroll
            for (int j = 0; j < (KC * DCH / 4) / 256; ++j)
                dstv[threadIdx.x + j * 256] = srcv[threadIdx.x + j * 256];
        } else {
            // padded case (lin2, 40 cols): scalar with zero fill
            for (int i = threadIdx.x; i < KC * DCH; i += 256) {
                int kr = i >> 7;
                int c  = i & (DCH - 1);
                bs[i] = (c < NCOLS) ? W[(kc + kr) * NCOLS + c] : 0.0f;
            }
        }
        __syncthreads();

        const float* Asel = (kc < DCH) ? A1 : A2;   // concat handling
        const int kbase = kc & (DCH - 1);

        if (!GUARD && (kc + KC < K)) {
            // prefetch next K-chunk of A for this wave's rows (inert hint)
            const float* An = ((kc + KC) < DCH) ? A1 : A2;
            __builtin_prefetch(An + (r0 + m_l) * DCH + ((kc + KC) & (DCH - 1)), 0, 0);
        }

        #pragma unroll
        for (int kk = 0; kk < KC; kk += 4) {
            // B 4x16 layout: VGPR v, lane L -> K = v + (L>>4)*2, N = L&15
            v2f b;
            b.x = bs[(kk + vk)     * DCH + ncol];
            b.y = bs[(kk + vk + 1) * DCH + ncol];

            // A 16x4 layout: lane L -> M = L&15; VGPR v -> K = v + (L>>4)*2
            v2f a0, a1, a2, a3;
            const int koff = kbase + kk + vk;
            if (GUARD) {
                v2f z = {0.f, 0.f};
                a0 = z; a1 = z; a2 = z; a3 = z;
                int r;
                r = r0      + m_l; if (r < M) a0 = *(const v2f*)(Asel + r * DCH + koff);
                r = r0 + 16 + m_l; if (r < M) a1 = *(const v2f*)(Asel + r * DCH + koff);
                r = r0 + 32 + m_l; if (r < M) a2 = *(const v2f*)(Asel + r * DCH + koff);
                r = r0 + 48 + m_l; if (r < M) a3 = *(const v2f*)(Asel + r * DCH + koff);
            } else {
                a0 = *(const v2f*)(Asel + (r0      + m_l) * DCH + koff);
                a1 = *(const v2f*)(Asel + (r0 + 16 + m_l) * DCH + koff);
                a2 = *(const v2f*)(Asel + (r0 + 32 + m_l) * DCH + koff);
                a3 = *(const v2f*)(Asel + (r0 + 48 + m_l) * DCH + koff);
            }

            acc[0] = __builtin_amdgcn_wmma_f32_16x16x4_f32(false, a0, false, b, (short)0, acc[0], false, false);
            acc[1] = __builtin_amdgcn_wmma_f32_16x16x4_f32(false, a1, false, b, (short)0, acc[1], false, false);
            acc[2] = __builtin_amdgcn_wmma_f32_16x16x4_f32(false, a2, false, b, (short)0, acc[2], false, false);
            acc[3] = __builtin_amdgcn_wmma_f32_16x16x4_f32(false, a3, false, b, (short)0, acc[3], false, false);
        }
        __syncthreads();
    }

    // C/D layout: VGPR i, lane L -> M = i + (L>>4)*8, N = L&15
    const int roff = (lane >> 4) << 3;
    float bval = (mode != 0 && ncol < NCOLS) ? bias[ncol] : 0.0f;
    #pragma unroll
    for (int s = 0; s < 4; ++s) {
        #pragma unroll
        for (int i = 0; i < 8; ++i) {
            int row = r0 + s * 16 + roff + i;
            if ((!GUARD || row < M) && ncol < NCOLS) {
                float v = acc[s][i] + bval;
                if (mode == 1) v = fmaxf(v, 0.0f);
                Out[(long long)row * ldout + ncol] = v;
            }
        }
    }
}

// ------------------------- edge scatter-aggregate ---------------------------
// One wave per edge; lane handles 4 consecutive channels (float4).
__global__ __launch_bounds__(256) void scatter_kernel(
    const float* __restrict__ t, const int* __restrict__ src,
    const int* __restrict__ dst, const float* __restrict__ dinv,
    float* __restrict__ agg, int E)
{
    long long idx = (long long)blockIdx.x * 256 + threadIdx.x;
    long long e = idx >> 5;
    if (e >= E) return;
    int l = ((int)idx & 31) << 2;          // channel base
    int s = src[e], d = dst[e];            // uniform across the wave
    float w = dinv[s] * dinv[d];
    const float4 v = *(const float4*)(t + (long long)s * DCH + l);
    float* p = agg + (long long)d * DCH + l;
    atomicAdd(p + 0, v.x * w);
    atomicAdd(p + 1, v.y * w);
    atomicAdd(p + 2, v.z * w);
    atomicAdd(p + 3, v.w * w);
}

// ---------------- GCN combine + BatchNorm statistics (pass 1) ---------------
// t <- agg + dinv^2 * t + bias   (in place), accumulate column sum / sumsq.
// Block covers 64 rows; thread (c, half) sums 32 rows of column c.
__global__ __launch_bounds__(256) void bnstat_kernel(
    float* __restrict__ t, const float* __restrict__ agg,
    const float* __restrict__ dinv, const float* __restrict__ bias,
    float* __restrict__ colsum, float* __restrict__ colsumsq, int M)
{
    __shared__ float s1[256];
    __shared__ float s2[256];
    int c    = threadIdx.x & (DCH - 1);
    int half = threadIdx.x >> 7;
    int rbeg = blockIdx.x * 64 + half * 32;
    float b = bias[c];
    float sum = 0.0f, sumsq = 0.0f;
    for (int j = 0; j < 32; ++j) {
        int r = rbeg + j;
        if (r < M) {
            float di = dinv[r];
            long long o = (long long)r * DCH + c;
            float v = agg[o] + di * di * t[o] + b;
            t[o] = v;
            sum += v;
            sumsq += v * v;
        }
    }
    s1[threadIdx.x] = sum;
    s2[threadIdx.x] = sumsq;
    __syncthreads();
    if (half == 0) {
        atomicAdd(&colsum[c],   s1[c] + s1[c + 128]);
        atomicAdd(&colsumsq[c], s2[c] + s2[c + 128]);
    }
}

// --------------------- BatchNorm finalize (tiny kernel) ---------------------
__global__ void bnfinal_kernel(const float* __restrict__ colsum,
                               const float* __restrict__ colsumsq,
                               const float* __restrict__ gamma,
                               const float* __restrict__ beta,
                               float* __restrict__ scale,
                               float* __restrict__ shift, float invM)
{
    int c = threadIdx.x;
    float mu   = colsum[c] * invM;
    float var  = colsumsq[c] * invM - mu * mu;   // biased variance
    float rstd = rsqrtf(var + 1e-5f);
    float sc   = gamma[c] * rstd;
    scale[c] = sc;
    shift[c] = beta[c] - mu * sc;
}

// ------------------------- BN apply + ReLU (pass 2) -------------------------
__global__ __launch_bounds__(256) void bnapply_kernel(
    const float* __restrict__ t, float* __restrict__ h,
    const float* __restrict__ scale, const float* __restrict__ shift,
    long long total)
{
    long long i = (long long)blockIdx.x * 256 + threadIdx.x;
    if (i < total) {
        int c = (int)(i & (DCH - 1));
        h[i] = fmaxf(t[i] * scale[c] + shift[c], 0.0f);
    }
}

// ------------------------------- launcher -----------------------------------

static inline void launch_gemm(const float* A1, const float* A2, const float* W,
                               const float* bias, float* Out, int M, int K,
                               int NCOLS, int ldout, int mode, hipStream_t stream)
{
    const int fullBlocks = M / 64;          // guard-free blocks
    const int tail = M - fullBlocks * 64;   // ragged rows
    if (fullBlocks > 0)
        gemm_wmma_kernel<false><<<fullBlocks, dim3(256), 0, stream>>>(
            A1, A2, W, bias, Out, M, K, NCOLS, ldout, mode, 0);
    if (tail > 0)
        gemm_wmma_kernel<true><<<1, dim3(256), 0, stream>>>(
            A1, A2, W, bias, Out, M, K, NCOLS, ldout, mode, fullBlocks);
}

extern "C" void kernel_launch(void* const* d_in, const int* in_sizes, int n_in,
                              void* d_out, int out_size, void* d_ws, size_t ws_size,
                              hipStream_t stream)
{
    const float* x      = (const float*)d_in[0];
    const float* fair   = (const float*)d_in[1];
    const int*   eidx   = (const int*)  d_in[2];
    const float* conv_w = (const float*)d_in[3];
    const float* conv_b = (const float*)d_in[4];
    const float* bn_g   = (const float*)d_in[5];
    const float* bn_b   = (const float*)d_in[6];
    const float* lin1_w = (const float*)d_in[7];
    const float* lin1_b = (const float*)d_in[8];
    const float* lin2_w = (const float*)d_in[9];
    const float* lin2_b = (const float*)d_in[10];

    const int M    = in_sizes[0] / DCH;     // nodes (100000)
    const int E    = in_sizes[2] / 2;       // directed edges (1.6M)
    const int OUTC = in_sizes[10];          // 40
    const int L    = in_sizes[4] / DCH;     // 3 conv layers

    const int* src = eidx;
    const int* dst = eidx + E;

    // workspace layout (floats)
    float* ws       = (float*)d_ws;
    float* dinv     = ws;                           // [M]  (also degree scratch)
    float* t        = dinv + M;                     // [M*128] gemm out / BN io
    float* agg      = t    + (size_t)M * DCH;       // [M*128] edge aggregate
    float* h        = agg  + (size_t)M * DCH;       // [M*128] layer activation
    float* colsum   = h    + (size_t)M * DCH;       // [128]
    float* colsumsq = colsum + DCH;                 // [128]
    float* bnscale  = colsumsq + DCH;               // [128]
    float* bnshift  = bnscale + DCH;                // [128]

    const dim3 b256(256);
    const long long totalMD = (long long)M * DCH;
    const int ewGrid   = (int)((totalMD + 255) / 256);
    const int scatGrid = (int)(((long long)E * 32 + 255) / 256);

    // degrees -> dinv = rsqrt(deg + 1)
    zero_kernel<<<512, b256, 0, stream>>>(dinv, (long long)M);
    deg_kernel<<<(E + 255) / 256, b256, 0, stream>>>(dst, dinv, E);
    dinv_kernel<<<(M + 255) / 256, b256, 0, stream>>>(dinv, M);

    const float* hin = x;
    for (int l = 0; l < L; ++l) {
        // t = hin @ conv_w[l]
        launch_gemm(hin, hin, conv_w + (size_t)l * DCH * DCH, nullptr,
                    t, M, DCH, DCH, DCH, 0, stream);
        zero_kernel<<<2048, b256, 0, stream>>>(agg, totalMD);
        zero_kernel<<<1, b256, 0, stream>>>(colsum, 2 * DCH);
        // agg[dst] += enorm * t[src]
        scatter_kernel<<<scatGrid, b256, 0, stream>>>(t, src, dst, dinv, agg, E);
        // t = agg + sloop*t + bias ; column stats
        bnstat_kernel<<<(M + 63) / 64, b256, 0, stream>>>(
            t, agg, dinv, conv_b + (size_t)l * DCH, colsum, colsumsq, M);
        bnfinal_kernel<<<1, dim3(DCH), 0, stream>>>(
            colsum, colsumsq, bn_g + (size_t)l * DCH, bn_b + (size_t)l * DCH,
            bnscale, bnshift, 1.0f / (float)M);
        // h = relu(bn(t))
        bnapply_kernel<<<ewGrid, b256, 0, stream>>>(t, h, bnscale, bnshift, totalMD);
        hin = h;
    }

    // t = relu([h, fair] @ lin1_w + lin1_b)   (K = 256, split across h / fair)
    launch_gemm(h, fair, lin1_w, lin1_b, t, M, 2 * DCH, DCH, DCH, 1, stream);
    // out = t @ lin2_w + lin2_b               (NCOLS = 40, zero-padded to 128)
    launch_gemm(t, t, lin2_w, lin2_b, (float*)d_out, M, DCH, OUTC, OUTC, 2, stream);
}